// ConcatLayer_57982058496361
// MI455X (gfx1250) — compile-verified
//
#include <hip/hip_runtime.h>

typedef float v4f __attribute__((ext_vector_type(4)));

namespace {

constexpr int kThreads        = 256;                     // 8 waves (wave32) per block
constexpr int kRowsPerBlock   = 256;                     // one row per thread
constexpr int kFloatsPerBlock = kRowsPerBlock * 9;       // 2304 floats = 9216 B
constexpr int kOutVec4PerBlk  = (kRowsPerBlock * 3) / 4; // 192 v4f stores per block

// Per-row logic, registers only, no dynamic indexing (everything -> v_cndmask).
__device__ __forceinline__ void compute_row(const float x[9], float o[3]) {
  int mi[3];
#pragma unroll
  for (int j = 0; j < 3; ++j) {
    const float a = x[3 * j + 0], b = x[3 * j + 1], c = x[3 * j + 2];
    const float m = fmaxf(a, fmaxf(b, c));
    const int ties = (int)(a == m) + (int)(b == m) + (int)(c == m);
    // first-occurrence argmax (strict > matches jnp.argmax tie rule)
    int idx = 0;
    float best = a;
    if (b > best) { best = b; idx = 1; }
    if (c > best) { idx = 2; }
    mi[j] = (ties > 1) ? 0 : (1 - idx);   // CONVERT = [1, 0, -1]
  }
  const int calc = ((mi[1] < 0) ? -mi[1] : mi[1]) * (mi[0] + mi[1] + mi[2]);

  float v[9];
#pragma unroll
  for (int j = 0; j < 3; ++j) {
    const bool keep = ((calc > 0) & (mi[j] > 0)) |
                      ((calc < 0) & (mi[j] < 0)) |
                      ((calc == 0) & (mi[j] == 0));
    v[3 * j + 0] = keep ? x[3 * j + 0] : 0.0f;
    v[3 * j + 1] = keep ? x[3 * j + 1] : 0.0f;
    v[3 * j + 2] = keep ? x[3 * j + 2] : 0.0f;
  }

  const int sel = (calc == 0) ? 1 : ((calc > 0) ? 0 : 2);
  float vals[3];
#pragma unroll
  for (int j = 0; j < 3; ++j)
    vals[j] = (sel == 0) ? v[3 * j + 0] : ((sel == 1) ? v[3 * j + 1] : v[3 * j + 2]);

  int win = 0;
  float bv = vals[0];
  if (vals[1] > bv) { bv = vals[1]; win = 1; }
  if (vals[2] > bv) { win = 2; }

  o[0] = (win == 0) ? v[0] : ((win == 1) ? v[3] : v[6]);
  o[1] = (win == 0) ? v[1] : ((win == 1) ? v[4] : v[7]);
  o[2] = (win == 0) ? v[2] : ((win == 1) ? v[5] : v[8]);
}

__global__ __launch_bounds__(kThreads) void triad_select_kernel(
    const float* __restrict__ in, float* __restrict__ out, long long nrows) {
  __shared__ float lds[kFloatsPerBlock];
  const int tid = threadIdx.x;
  const long long row_base = (long long)blockIdx.x * kRowsPerBlock;

  if (row_base + kRowsPerBlock <= nrows) {
    // ---- Phase 1: coalesced NT b128 global -> LDS staging.
    // 576 v4f per block = 256 + 256 + 64. Issue ALL loads first (3 in flight
    // per wave), then the ds_store_b128s, so the wave isn't stuck at
    // loadcnt==0 between each load/store pair.
    const v4f* __restrict__ in4 = reinterpret_cast<const v4f*>(in + row_base * 9);
    v4f* lds4 = reinterpret_cast<v4f*>(lds);

    const bool third = (tid < 64);
    v4f r0 = __builtin_nontemporal_load(in4 + tid);
    v4f r1 = __builtin_nontemporal_load(in4 + tid + 256);
    v4f r2 = {};
    if (third) r2 = __builtin_nontemporal_load(in4 + tid + 512);

    lds4[tid]       = r0;
    lds4[tid + 256] = r1;
    if (third) lds4[tid + 512] = r2;
    __syncthreads();

    // ---- Phase 2: each lane reads its row from LDS (stride 9: bank-conflict-free) ----
    float x[9];
#pragma unroll
    for (int k = 0; k < 9; ++k) x[k] = lds[tid * 9 + k];

    float o[3];
    compute_row(x, o);

    // ---- Phase 3: stage 3-float outputs, then coalesced NT b128 stores ----
    __syncthreads();  // all staging reads done before LDS reuse
    lds[tid * 3 + 0] = o[0];
    lds[tid * 3 + 1] = o[1];
    lds[tid * 3 + 2] = o[2];
    __syncthreads();

    v4f* __restrict__ out4 = reinterpret_cast<v4f*>(out + row_base * 3);
    if (tid < kOutVec4PerBlk)
      __builtin_nontemporal_store(lds4[tid], out4 + tid);
  } else {
    // Tail block (not hit for N = 8388608, kept for generality).
    const long long row = row_base + tid;
    if (row < nrows) {
      float x[9];
#pragma unroll
      for (int k = 0; k < 9; ++k) x[k] = in[row * 9 + k];
      float o[3];
      compute_row(x, o);
      out[row * 3 + 0] = o[0];
      out[row * 3 + 1] = o[1];
      out[row * 3 + 2] = o[2];
    }
  }
}

}  // namespace

extern "C" void kernel_launch(void* const* d_in, const int* in_sizes, int n_in,
                              void* d_out, int out_size, void* d_ws, size_t ws_size,
                              hipStream_t stream) {
  const float* in = (const float*)d_in[0];
  float* out = (float*)d_out;
  const long long nrows = (long long)in_sizes[0] / 9;   // 8,388,608
  const int blocks = (int)((nrows + kRowsPerBlock - 1) / kRowsPerBlock);  // 32768
  triad_select_kernel<<<blocks, kThreads, 0, stream>>>(in, out, nrows);
}